// GrGCNLayerGeomstats_83545703842381
// MI455X (gfx1250) — compile-verified
//
#include <hip/hip_runtime.h>

#define BB 8
#define NN 2048
#define DD 32
#define PIN 8
#define POUT 4
#define JJ (DD * POUT)  // 128

typedef __attribute__((ext_vector_type(2))) float v2f;
typedef __attribute__((ext_vector_type(4))) float v4f;
typedef __attribute__((ext_vector_type(8))) float v8f;

// ---------------------------------------------------------------------------
// Kernel 1: Mn = U @ W_neigh ; Msf = U @ W_self + bias
// one thread per (b,n,d) point: 8 inputs -> 4+4 outputs
// ---------------------------------------------------------------------------
__global__ __launch_bounds__(256) void prep_kernel(
    const float* __restrict__ U, const float* __restrict__ Wself,
    const float* __restrict__ Wneigh, const float* __restrict__ bias,
    float* __restrict__ Mn, float* __restrict__ Msf) {
  int t = blockIdx.x * blockDim.x + threadIdx.x;  // 0 .. B*N*D-1
  const float* u = U + (long)t * PIN;
  float ui[PIN];
#pragma unroll
  for (int i = 0; i < PIN; i++) ui[i] = u[i];
  float mn[POUT], ms[POUT];
#pragma unroll
  for (int o = 0; o < POUT; o++) {
    mn[o] = 0.0f;
    ms[o] = bias[o];
  }
#pragma unroll
  for (int i = 0; i < PIN; i++) {
#pragma unroll
    for (int o = 0; o < POUT; o++) {
      ms[o] = fmaf(ui[i], Wself[i * POUT + o], ms[o]);
      mn[o] = fmaf(ui[i], Wneigh[i * POUT + o], mn[o]);
    }
  }
  long off = (long)t * POUT;
  v4f vn = {mn[0], mn[1], mn[2], mn[3]};
  v4f vs = {ms[0], ms[1], ms[2], ms[3]};
  *(v4f*)(Mn + off) = vn;
  *(v4f*)(Msf + off) = vs;
}

// ---------------------------------------------------------------------------
// Kernel 2: Y = relu(A @ Mn + Msf + bias)   via V_WMMA_F32_16X16X4_F32
// block = 256 thr (8 waves); wave w does a 16x32 tile (two accumulators),
// block covers 128 rows x 32 cols. Each A fragment feeds TWO wmma ops.
//
// B panel staged in LDS in FRAGMENT-READY interleaved layout:
//   element (k, col) at lds[(k>>1)*PSTRIDE + col*2 + (k&1)], PSTRIDE = 96.
// A B-fragment is then one aligned ds_load_b64: {(k,col),(k+1,col)} directly
// in an even VGPR pair (VGPR0=K, VGPR1=K+1), no regrouping movs.
// PSTRIDE = 96 == 32 (mod 64 banks) => the two lane-halves (pairs p, p+1)
// occupy disjoint 32-bank ranges: conflict-free.
//
// A fragment layout (ISA 7.12.2, 32-bit A 16x4): lane&15 = M row,
//   lanes 0-15: VGPR0=K, VGPR1=K+1 ; lanes 16-31: VGPR0=K+2, VGPR1=K+3.
// ---------------------------------------------------------------------------
__global__ __launch_bounds__(256) void gemm_kernel(
    const float* __restrict__ A, const float* __restrict__ Mn,
    const float* __restrict__ Msf, const float* __restrict__ bias,
    float* __restrict__ Y) {
  const int KB = 128;       // K-slab
  const int PSTRIDE = 96;   // floats per row-pair in LDS (64 data + 32 pad)
  __shared__ float ldsB[(KB / 2) * PSTRIDE];  // 24 KB

  const int tid = threadIdx.x;
  const int lane = tid & 31;
  const int wave = tid >> 5;  // 0..7
  const int b = blockIdx.z;
  const int row0 = blockIdx.x * 128 + wave * 16;
  const int col0 = blockIdx.y * 32;
  const int mlane = lane & 15;
  const int half = lane >> 4;
  const int koff = half * 2;

  const float* Arow = A + ((long)b * NN + row0 + mlane) * NN;
  const float* MnB = Mn + (long)b * NN * JJ;

  const int lr = tid >> 5;  // loader row 0..7 (one 128B row per wave)
  const int lc = tid & 31;  // loader col 0..31

  v8f acc0 = {};
  v8f acc1 = {};
  for (int kb = 0; kb < NN; kb += KB) {
    // stage Mn[kb..kb+KB) x [col0..col0+32) into LDS, interleaving row pairs
#pragma unroll
    for (int rp = 0; rp < KB; rp += 8) {
      int r = rp + lr;
      ldsB[(r >> 1) * PSTRIDE + lc * 2 + (r & 1)] =
          MnB[(long)(kb + r) * JJ + col0 + lc];
    }
    __syncthreads();
    if (kb + KB < NN) {
      __builtin_prefetch(Arow + kb + KB, 0, 0);  // global_prefetch_b8
    }
#pragma unroll 8
    for (int kk = 0; kk < KB; kk += 4) {
      v2f afrag = *(const v2f*)(Arow + kb + kk + koff);
      int p = (kk + koff) >> 1;  // row-pair index for this lane-half
      const float* pb = ldsB + p * PSTRIDE + mlane * 2;
      v2f b0 = *(const v2f*)(pb);       // {(k,col),(k+1,col)}
      v2f b1 = *(const v2f*)(pb + 32);  // col + 16
      acc0 = __builtin_amdgcn_wmma_f32_16x16x4_f32(
          false, afrag, false, b0, (short)0, acc0, false, false);
      acc1 = __builtin_amdgcn_wmma_f32_16x16x4_f32(
          false, afrag, false, b1, (short)0, acc1, false, false);
    }
    __syncthreads();
  }

  // Epilogue: C/D layout -> VGPR r: lanes 0-15 M=r, lanes 16-31 M=r+8
#pragma unroll
  for (int r = 0; r < 8; r++) {
    int row = row0 + r + 8 * half;
    int col = col0 + mlane;
    long idx = ((long)b * NN + row) * JJ + col;
    float y0 = acc0[r] + Msf[idx] + bias[col & 3];
    Y[idx] = fmaxf(y0, 0.0f);
    long idx1 = idx + 16;
    float y1 = acc1[r] + Msf[idx1] + bias[(col + 16) & 3];
    Y[idx1] = fmaxf(y1, 0.0f);
  }
}

// ---------------------------------------------------------------------------
// Kernel 3: in-place Stiefel (polar) projection of each 32x4 block of Y.
// One wave per matrix; lane d holds row d. Gram via butterfly shuffles,
// coupled Newton-Schulz for Gn^{-1/2} (Gn = G/tr(G)), then Y <- Y G^{-1/2}.
// ---------------------------------------------------------------------------
__global__ __launch_bounds__(256) void stiefel_kernel(float* __restrict__ Y) {
  int gtid = blockIdx.x * blockDim.x + threadIdx.x;
  int wid = gtid >> 5;   // matrix index 0 .. B*N-1
  int lane = gtid & 31;  // row index d

  float* ym = Y + (long)wid * JJ;
  v4f yr = *(const v4f*)(ym + lane * 4);
  float y[4] = {yr.x, yr.y, yr.z, yr.w};

  // Gram G = Y^T Y (all lanes end with the full 4x4)
  float G[4][4];
#pragma unroll
  for (int i = 0; i < 4; i++) {
#pragma unroll
    for (int j = i; j < 4; j++) {
      float p = y[i] * y[j];
#pragma unroll
      for (int off = 16; off >= 1; off >>= 1) p += __shfl_xor(p, off, 32);
      G[i][j] = p;
      G[j][i] = p;
    }
  }
  float tr = G[0][0] + G[1][1] + G[2][2] + G[3][3];

  float out[4] = {y[0], y[1], y[2], y[3]};  // fallback: Y ~ 0 -> pass through
  if (tr > 1e-20f) {
    float inv_t = 1.0f / tr;
    float Yk[4][4], Zk[4][4];
#pragma unroll
    for (int i = 0; i < 4; i++) {
#pragma unroll
      for (int j = 0; j < 4; j++) {
        Yk[i][j] = G[i][j] * inv_t;
        Zk[i][j] = (i == j) ? 1.0f : 0.0f;
      }
    }
    // Coupled Newton-Schulz: T = 0.5(3I - Z Y); Y <- Y T; Z <- T Z
#pragma unroll 1
    for (int it = 0; it < 24; it++) {
      float Tm[4][4];
#pragma unroll
      for (int i = 0; i < 4; i++) {
#pragma unroll
        for (int j = 0; j < 4; j++) {
          float s = 0.0f;
#pragma unroll
          for (int k = 0; k < 4; k++) s = fmaf(Zk[i][k], Yk[k][j], s);
          Tm[i][j] = (i == j) ? (1.5f - 0.5f * s) : (-0.5f * s);
        }
      }
      float Yn[4][4], Zn[4][4];
#pragma unroll
      for (int i = 0; i < 4; i++) {
#pragma unroll
        for (int j = 0; j < 4; j++) {
          float sy = 0.0f, sz = 0.0f;
#pragma unroll
          for (int k = 0; k < 4; k++) {
            sy = fmaf(Yk[i][k], Tm[k][j], sy);
            sz = fmaf(Tm[i][k], Zk[k][j], sz);
          }
          Yn[i][j] = sy;
          Zn[i][j] = sz;
        }
      }
#pragma unroll
      for (int i = 0; i < 4; i++)
#pragma unroll
        for (int j = 0; j < 4; j++) {
          Yk[i][j] = Yn[i][j];
          Zk[i][j] = Zn[i][j];
        }
    }
    float s = rsqrtf(tr);  // G^{-1/2} = Zk / sqrt(tr)
#pragma unroll
    for (int o = 0; o < 4; o++) {
      float a = 0.0f;
#pragma unroll
      for (int j = 0; j < 4; j++) a = fmaf(y[j], Zk[j][o], a);
      out[o] = a * s;
    }
  }
  v4f vo = {out[0], out[1], out[2], out[3]};
  *(v4f*)(ym + lane * 4) = vo;
}

// ---------------------------------------------------------------------------
extern "C" void kernel_launch(void* const* d_in, const int* in_sizes, int n_in,
                              void* d_out, int out_size, void* d_ws,
                              size_t ws_size, hipStream_t stream) {
  const float* U = (const float*)d_in[0];       // (B,N,D,PIN)
  const float* A = (const float*)d_in[1];       // (B,N,N)
  const float* Wself = (const float*)d_in[2];   // (PIN,POUT)
  const float* Wneigh = (const float*)d_in[3];  // (PIN,POUT)
  const float* bias = (const float*)d_in[4];    // (POUT)
  float* Y = (float*)d_out;                     // (B,N,D,POUT) == (B,N,128)

  float* Mn = (float*)d_ws;              // 8 MB
  float* Msf = Mn + (long)BB * NN * JJ;  // 8 MB

  // K1: per-point projections
  {
    int total = BB * NN * DD;  // 524288
    prep_kernel<<<total / 256, 256, 0, stream>>>(U, Wself, Wneigh, bias, Mn,
                                                 Msf);
  }
  // K2: batched fp32 WMMA GEMM + relu epilogue -> Y (in d_out)
  {
    dim3 grid(NN / 128, JJ / 32, BB);  // (16, 4, 8)
    gemm_kernel<<<grid, 256, 0, stream>>>(A, Mn, Msf, bias, Y);
  }
  // K3: in-place polar projection, one wave per 32x4 matrix
  {
    int total = BB * NN * 32;  // one lane per row, 32 lanes per matrix
    stiefel_kernel<<<total / 256, 256, 0, stream>>>(Y);
  }
}